// RGU_Layer_34978213658992
// MI455X (gfx1250) — compile-verified
//
#include <hip/hip_runtime.h>
#include <math.h>

// GRU scan for MI455X (gfx1250, wave32).
// x:[512,64,1024] f32, W_ih:[1024,3072], W_hh:[1024,3072], biases:[3072].
// out = concat(sequence [512,64,1024], h_last [64,1024]) flat f32.
//
// Latency-bound serial recurrence -> one persistent kernel, grid barrier per
// step, weights L2-resident (24 MB << 192 MB L2), fp32 kept via
// V_WMMA_F32_16X16X4_F32. x-projection fused as first half of K=2048.
// LDS layouts tuned so every WMMA fragment is a single aligned b64 LDS load.

#define T_LEN 512
#define BATCH 64
#define D_HID 1024
#define N3    3072
#define NBLK  32      // 1024 hidden cols / 32 per block
#define LDA   36      // A tile row stride (floats): 16B-aligned, conflict-free
#define LDBP  224     // B tile pair stride (floats): upper/lower bank split

typedef float v2f __attribute__((ext_vector_type(2)));
typedef float v8f __attribute__((ext_vector_type(8)));

__device__ __forceinline__ v8f wmma4(v2f a, v2f b, v8f c) {
  // D = A(16x4 f32) * B(4x16 f32) + C(16x16 f32)
  return __builtin_amdgcn_wmma_f32_16x16x4_f32(
      /*neg_a=*/false, a, /*neg_b=*/false, b,
      /*c_mod=*/(short)0, c, /*reuse_a=*/false, /*reuse_b=*/false);
}

__global__ void sync_init_kernel(unsigned int* s) { *s = 0u; }

__global__ __launch_bounds__(256) void gru_scan_kernel(
    const float* __restrict__ x,
    const float* __restrict__ h0,
    const float* __restrict__ W_ih,
    const float* __restrict__ W_hh,
    const float* __restrict__ b_ih,
    const float* __restrict__ b_hh,
    float* __restrict__ seq,
    float* __restrict__ h_last,
    unsigned int* __restrict__ sync)
{
  __shared__ float sA[BATCH * LDA];   // 64 rows x 32 K (padded)     9.2 KB
  __shared__ float sB[16 * LDBP];     // 16 K-pairs x 96 cols x2     14.3 KB

  const int tid   = threadIdx.x;
  const int j0    = blockIdx.x * 32;        // hidden-column base for this block
  const int wv    = tid >> 5;               // wave id 0..7
  const int lane  = tid & 31;
  const int laneM = lane & 15;
  const int hi    = lane >> 4;
  const int m0    = (wv >> 1) << 4;         // 16-row tile base (4 tiles)
  const int c0    = (wv & 1) << 4;          // 16-col tile base within 32 cols
  const int col   = j0 + c0 + laneM;        // this lane's hidden column

  const float br  = b_ih[col]            + b_hh[col];
  const float bz  = b_ih[D_HID + col]    + b_hh[D_HID + col];
  const float bxn = b_ih[2 * D_HID + col];
  const float bhn = b_hh[2 * D_HID + col];

  for (int t = 0; t < T_LEN; ++t) {
    const float* xt = x + (size_t)t * (BATCH * D_HID);
    const float* hp = (t == 0) ? h0 : (seq + (size_t)(t - 1) * (BATCH * D_HID));
    float*       ho = seq + (size_t)t * (BATCH * D_HID);

    v8f acc_r  = {0, 0, 0, 0, 0, 0, 0, 0};
    v8f acc_z  = {0, 0, 0, 0, 0, 0, 0, 0};
    v8f acc_xn = {0, 0, 0, 0, 0, 0, 0, 0};
    v8f acc_hn = {0, 0, 0, 0, 0, 0, 0, 0};

    // One half of the fused K=2048 accumulation (K=1024 with given A/B source).
    auto half_gemm = [&](const float* __restrict__ Asrc,
                         const float* __restrict__ Bsrc,
                         v8f& acc_n) {
      for (int kb = 0; kb < 32; ++kb) {
        const int goff = kb * 32;
        // Stage A tile: 64 rows x 32 K, b128 loads/stores (2 per thread).
        #pragma unroll
        for (int rr = 0; rr < 2; ++rr) {
          const int e   = rr * 256 + tid;       // 0..511
          const int row = e >> 3;
          const int q   = (e & 7) * 4;
          *(float4*)&sA[row * LDA + q] =
              *(const float4*)&Asrc[(size_t)row * D_HID + goff + q];
        }
        // Stage B tile: 32 K x 96 cols (r|z|n), K-pair-interleaved in LDS so a
        // fragment {B[k][c], B[k+1][c]} is one aligned b64 load.
        #pragma unroll
        for (int rr = 0; rr < 3; ++rr) {
          const int e    = rr * 256 + tid;      // 0..767 float4s
          const int krow = e / 24;
          const int q4   = (e % 24) * 4;        // col 0..92 (stays in one gate)
          const int gate = q4 >> 5;
          const int cj   = q4 & 31;
          const float4 v = *(const float4*)
              &Bsrc[(size_t)(goff + krow) * N3 + gate * D_HID + j0 + cj];
          float* dst = &sB[(krow >> 1) * LDBP + q4 * 2 + (krow & 1)];
          dst[0] = v.x; dst[2] = v.y; dst[4] = v.z; dst[6] = v.w;
        }
        // Prefetch next K-block's cachelines into L2/L0 while we compute.
        if (kb < 31) {
          if (tid < 96) {       // 32 rows x 3 gate strips (128B each)
            __builtin_prefetch(&Bsrc[(size_t)(goff + 32 + (tid & 31)) * N3 +
                                     (tid >> 5) * D_HID + j0], 0, 0);
          } else if (tid < 160) { // 64 A rows
            __builtin_prefetch(&Asrc[(size_t)(tid - 96) * D_HID + goff + 32],
                               0, 0);
          }
        }
        __syncthreads();
        #pragma unroll
        for (int ks = 0; ks < 8; ++ks) {
          const int k0 = ks * 4;
          // A frag: lanes 0-15 K={0,1}, lanes 16-31 K={2,3} (b64 from LDS).
          v2f a = *(const v2f*)&sA[(m0 + laneM) * LDA + k0 + 2 * hi];
          const float* bp = &sB[((k0 >> 1) + hi) * LDBP + (c0 + laneM) * 2];
          v2f b0 = *(const v2f*)&bp[0];     // gate r cols
          acc_r = wmma4(a, b0, acc_r);
          v2f b1 = *(const v2f*)&bp[64];    // gate z cols (+32 cols * 2)
          acc_z = wmma4(a, b1, acc_z);
          v2f b2 = *(const v2f*)&bp[128];   // gate n cols
          acc_n = wmma4(a, b2, acc_n);
        }
        __syncthreads();
      }
    };

    half_gemm(xt, W_ih, acc_xn);   // input-side projection (fused)
    half_gemm(hp, W_hh, acc_hn);   // hidden-side projection

    // Gate math + state update. C layout: VGPR i -> row m0 + i + 8*hi, col laneM.
    #pragma unroll
    for (int i = 0; i < 8; ++i) {
      const int row = m0 + i + 8 * hi;
      const float r = 1.0f / (1.0f + __expf(-(acc_r[i] + br)));
      const float z = 1.0f / (1.0f + __expf(-(acc_z[i] + bz)));
      const float u = acc_xn[i] + bxn + r * (acc_hn[i] + bhn);
      const float n = 1.0f - 2.0f / (__expf(2.0f * u) + 1.0f);  // tanh, sat-safe
      const float hprev = hp[(size_t)row * D_HID + col];
      const float hnew  = (1.0f - z) * n + z * hprev;
      ho[(size_t)row * D_HID + col] = hnew;
      if (t == T_LEN - 1) h_last[(size_t)row * D_HID + col] = hnew;
    }

    // Device-scope grid barrier between steps (32 co-resident blocks).
    __threadfence();
    __syncthreads();
    if (tid == 0) {
      atomicAdd(sync, 1u);
      const unsigned int target = (unsigned int)NBLK * (unsigned int)(t + 1);
      while (__hip_atomic_load(sync, __ATOMIC_ACQUIRE,
                               __HIP_MEMORY_SCOPE_AGENT) < target) {
        __builtin_amdgcn_s_sleep(1);
      }
    }
    __syncthreads();
  }
}

extern "C" void kernel_launch(void* const* d_in, const int* in_sizes, int n_in,
                              void* d_out, int out_size, void* d_ws, size_t ws_size,
                              hipStream_t stream) {
  (void)in_sizes; (void)n_in; (void)out_size; (void)ws_size;
  const float* x    = (const float*)d_in[0];
  const float* h0   = (const float*)d_in[1];
  const float* W_ih = (const float*)d_in[2];
  const float* W_hh = (const float*)d_in[3];
  const float* b_ih = (const float*)d_in[4];
  const float* b_hh = (const float*)d_in[5];

  float* seq    = (float*)d_out;
  float* h_last = seq + (size_t)T_LEN * BATCH * D_HID;
  unsigned int* sync = (unsigned int*)d_ws;

  // ws is poisoned once and never re-poisoned: reset the barrier counter
  // every call so replays are deterministic.
  sync_init_kernel<<<1, 1, 0, stream>>>(sync);
  gru_scan_kernel<<<NBLK, 256, 0, stream>>>(x, h0, W_ih, W_hh, b_ih, b_hh,
                                            seq, h_last, sync);
}